// LOSS_FG_BG_34909494182413
// MI455X (gfx1250) — compile-verified
//
#include <hip/hip_runtime.h>

typedef __attribute__((ext_vector_type(2))) float v2f;
typedef __attribute__((ext_vector_type(8))) float v8f;

#define NPIX  10000
#define NV4   2500
#define BATCH 4096

// One block per sample. Streams Pred/GT once with B128 loads, accumulates the
// six masked moments, folds across the block, and emits per-sample fg/bg
// variance. Final cross-wave fold uses V_WMMA_F32_16X16X4_F32 with an all-ones
// B matrix: D[q][n] = sum_k A[q][k], q = moment index.
__global__ __launch_bounds__(256) void sample_moments_kernel(
    const float* __restrict__ pred, const float* __restrict__ gt,
    float* __restrict__ ws_fg, float* __restrict__ ws_bg)
{
    const int b   = blockIdx.x;
    const int tid = threadIdx.x;
    const float4* p4 = (const float4*)(pred + (size_t)b * NPIX);
    const float4* g4 = (const float4*)(gt   + (size_t)b * NPIX);

    float fs = 0.f, fq = 0.f, fn = 0.f;   // fg: sum, sumsq, count
    float bs = 0.f, bq = 0.f, bn = 0.f;   // bg: sum, sumsq, count

    for (int v = tid; v < NV4; v += 256) {
        float4 p = p4[v];
        float4 g = g4[v];
        float pe[4] = {p.x, p.y, p.z, p.w};
        float ge[4] = {g.x, g.y, g.z, g.w};
#pragma unroll
        for (int e = 0; e < 4; ++e) {
            float x  = pe[e];
            bool nz  = (x != 0.0f);            // matches (Pred*mask != 0)
            bool fg  = (ge[e] > 0.5f);
            bool bg  = (ge[e] < 0.5f);
            float xf = fg ? x : 0.0f;
            float xb = bg ? x : 0.0f;
            fs += xf;  fq += xf * xf;  fn += (fg && nz) ? 1.0f : 0.0f;
            bs += xb;  bq += xb * xb;  bn += (bg && nz) ? 1.0f : 0.0f;
        }
    }

    // wave32 fold (gfx1250 is wave32-only)
#pragma unroll
    for (int off = 16; off >= 1; off >>= 1) {
        fs += __shfl_xor(fs, off, 32);
        fq += __shfl_xor(fq, off, 32);
        fn += __shfl_xor(fn, off, 32);
        bs += __shfl_xor(bs, off, 32);
        bq += __shfl_xor(bq, off, 32);
        bn += __shfl_xor(bn, off, 32);
    }

    __shared__ float P[6][8];                  // [moment][wave]
    const int wave = tid >> 5;
    if ((tid & 31) == 0) {
        P[0][wave] = fs; P[1][wave] = fq; P[2][wave] = fn;
        P[3][wave] = bs; P[4][wave] = bq; P[5][wave] = bn;
    }
    __syncthreads();

    if (tid < 32) {                            // wave 0 only: EXEC all-1s
        // A is 16x4 f32: lanes 0-15 hold K=0,1 (VGPR0,1); lanes 16-31 hold K=2,3.
        // Pre-add partial pairs so A[q][k] = P[q][k] + P[q][k+4], k=0..3.
        const int m  = tid & 15;               // matrix row = moment index
        const int k0 = (tid >> 4) << 1;        // 0 or 2
        float a0 = 0.0f, a1 = 0.0f;
        if (m < 6) {
            a0 = P[m][k0]     + P[m][k0 + 4];
            a1 = P[m][k0 + 1] + P[m][k0 + 5];
        }
        v2f A; A[0] = a0;   A[1] = a1;
        v2f B; B[0] = 1.0f; B[1] = 1.0f;       // all-ones B: layout-agnostic
        v8f C = {};
        v8f D = __builtin_amdgcn_wmma_f32_16x16x4_f32(
            /*neg_a=*/false, A, /*neg_b=*/false, B,
            /*c_mod=*/(short)0, C, /*reuse_a=*/false, /*reuse_b=*/false);
        // D layout: VGPR r, lanes 0-15 -> D[M=r][N=lane]; lane 0 sees rows 0..7.
        if (tid == 0) {
            float Fs = D[0], Fq = D[1], Fn = D[2];
            float Bs = D[3], Bq = D[4], Bn = D[5];
            ws_fg[b] = (Fq - Fs * Fs / Fn) / (Fn - 1.0f);
            ws_bg[b] = (Bq - Bs * Bs / Bn) / (Bn - 1.0f);
        }
    }
}

// 4096 -> 1 mean for each of fg/bg. Deterministic tree reduction, no atomics.
__global__ __launch_bounds__(256) void final_mean_kernel(
    const float* __restrict__ ws_fg, const float* __restrict__ ws_bg,
    float* __restrict__ out)
{
    __shared__ float sf[256], sb[256];
    const int tid = threadIdx.x;
    float f = 0.f, g = 0.f;
    for (int i = tid; i < BATCH; i += 256) { f += ws_fg[i]; g += ws_bg[i]; }
    sf[tid] = f; sb[tid] = g;
    __syncthreads();
    for (int s = 128; s > 0; s >>= 1) {
        if (tid < s) { sf[tid] += sf[tid + s]; sb[tid] += sb[tid + s]; }
        __syncthreads();
    }
    if (tid == 0) {
        out[0] = sf[0] * (1.0f / BATCH);
        out[1] = sb[0] * (1.0f / BATCH);
    }
}

extern "C" void kernel_launch(void* const* d_in, const int* in_sizes, int n_in,
                              void* d_out, int out_size, void* d_ws, size_t ws_size,
                              hipStream_t stream) {
    const float* pred = (const float*)d_in[0];   // [4096,1,100,100] f32
    const float* gt   = (const float*)d_in[1];   // [4096,1,100,100] f32
    float* ws  = (float*)d_ws;                   // 2*4096 f32 = 32 KB scratch
    float* out = (float*)d_out;                  // {fg_var, bg_var}

    sample_moments_kernel<<<BATCH, 256, 0, stream>>>(pred, gt, ws, ws + BATCH);
    final_mean_kernel<<<1, 256, 0, stream>>>(ws, ws + BATCH, out);
}